// StructureTensorEffect_55929064128865
// MI455X (gfx1250) — compile-verified
//
#include <hip/hip_runtime.h>
#include <cstdint>
#include <cstddef>

// ---------------------------------------------------------------------------
// StructureTensorEffect for MI455X (gfx1250).
// Bandwidth-bound stencil: 96MB min traffic @ 23.3TB/s ~ 4.1us floor.
// CDNA5 path used: async global->LDS staging (ASYNCcnt) + wave32 LDS stencil.
// WMMA intentionally NOT used: the only contraction is a length-8 tap sum with
// per-batch scalar weights, folded into the bilinear FMAs.
// ---------------------------------------------------------------------------

namespace {

constexpr int Wd = 1024, Hd = 1024, Bd = 4, Cd = 3;
constexpr int TX = 64, TY = 8, HALO = 3;
constexpr int LW = TX + 2 * HALO;        // 70
constexpr int LH = TY + 2 * HALO;        // 14
constexpr int LSTR = 72;                 // padded row stride (floats)
constexpr int PLANE = LSTR * LH;         // 1008 floats / channel
constexpr int NTHREADS = 256;            // 8 waves (wave32)
constexpr int STAGE_TOTAL = Cd * LH * LW; // 2940 elements

#if defined(__HIP_DEVICE_COMPILE__) && __has_builtin(__builtin_amdgcn_global_load_async_to_lds_b32)
#define ST_ASYNC_BUILTIN 1
#else
#define ST_ASYNC_BUILTIN 0
#endif
#if defined(__HIP_DEVICE_COMPILE__) && __has_builtin(__builtin_amdgcn_s_wait_asynccnt)
#define ST_WAIT_BUILTIN 1
#else
#define ST_WAIT_BUILTIN 0
#endif

typedef __attribute__((address_space(1))) int g1_int;   // global int
typedef __attribute__((address_space(3))) int l3_int;   // LDS int

// One 4-byte async copy global->LDS (per-lane addresses, so halo clamping is
// done on the *global* address; OOB never happens).
__device__ __forceinline__ void stage_b32(const float* __restrict__ src,
                                          float* dst_lds_generic)
{
#if defined(__HIP_DEVICE_COMPILE__)
    // builtin signature (from hipcc diagnostic): arg0 is AS1 int*, non-const
    g1_int* g = (g1_int*)(uintptr_t)src;
    // low 32 bits of a generic LDS pointer are the LDS byte offset
    l3_int* l = (l3_int*)(uint32_t)(uintptr_t)dst_lds_generic;
#if ST_ASYNC_BUILTIN
    __builtin_amdgcn_global_load_async_to_lds_b32(g, l, 0, 0);
#else
    uint32_t laddr = (uint32_t)(uintptr_t)dst_lds_generic;
    uint64_t gaddr = (uint64_t)(uintptr_t)src;
    asm volatile("global_load_async_to_lds_b32 %0, %1, off"
                 :: "v"(laddr), "v"(gaddr) : "memory");
#endif
#else
    (void)src; (void)dst_lds_generic;
#endif
}

__device__ __forceinline__ void wait_async_zero()
{
#if defined(__HIP_DEVICE_COMPILE__)
#if ST_WAIT_BUILTIN
    __builtin_amdgcn_s_wait_asynccnt(0);
#else
    asm volatile("s_wait_asynccnt 0" ::: "memory");
#endif
#endif
}

__global__ __launch_bounds__(NTHREADS)
void structure_tensor_kernel(const float* __restrict__ x,
                             const float* __restrict__ sigma_arr,
                             float* __restrict__ out)
{
    __shared__ float tile[Cd * PLANE];   // 12096 B

    const int tx   = threadIdx.x;        // 0..63
    const int tyq  = threadIdx.y;        // 0..3  (each thread does rows tyq, tyq+4)
    const int tid  = tyq * TX + tx;
    const int tileX = blockIdx.x * TX;
    const int tileY = blockIdx.y * TY;
    const int b     = blockIdx.z;

    const float* __restrict__ xin = x + (size_t)b * Cd * Hd * Wd;

    // ---- Stage input tile + clamped halo into LDS via async loads ----------
    for (int i = tid; i < STAGE_TOTAL; i += NTHREADS) {
        const int c   = i / (LH * LW);
        const int rem = i - c * (LH * LW);
        const int r   = rem / LW;
        const int col = rem - r * LW;
        int gy = tileY - HALO + r;
        gy = gy < 0 ? 0 : (gy > Hd - 1 ? Hd - 1 : gy);
        int gx = tileX - HALO + col;
        gx = gx < 0 ? 0 : (gx > Wd - 1 ? Wd - 1 : gx);
        stage_b32(xin + ((size_t)c * Hd + gy) * Wd + gx,
                  &tile[c * PLANE + r * LSTR + col]);
    }
    wait_async_zero();
    __syncthreads();

    // ---- Per-batch tap constants (sigma uniform across block) --------------
    // offsets are d*sigma for d in {-1,0,+1}: only 3 floor/frac pairs needed.
    const float sg = sigma_arr[b];
    int   ioi[3];
    float frf[3];
    {
        const float em = -sg; const float fm = floorf(em);
        ioi[0] = (int)fm; frf[0] = em - fm;
        ioi[1] = 0;       frf[1] = 0.0f;
        const float ep =  sg; const float fp = floorf(ep);
        ioi[2] = (int)fp; frf[2] = ep - fp;
    }

    constexpr int   OIU[8] = {0, 0, 0, 1, 1, 2, 2, 2};   // OFF_U + 1
    constexpr int   OIV[8] = {0, 1, 2, 0, 2, 0, 1, 2};   // OFF_V + 1
    constexpr float KUc[8] = {-0.25f, -0.5f, -0.25f, 0.0f, 0.0f, 0.25f, 0.5f, 0.25f};
    constexpr float KVc[8] = {-0.25f,  0.0f,  0.25f, -0.5f, 0.5f, -0.25f, 0.0f, 0.25f};
    constexpr float LUM[3] = {100.0f, 1.0f, 1.0f};

    const int lx = tx;

    #pragma unroll
    for (int ry = 0; ry < 2; ++ry) {
        const int ly = tyq + ry * 4;     // 0..7

        float su[3] = {0.0f, 0.0f, 0.0f};
        float sv[3] = {0.0f, 0.0f, 0.0f};

        #pragma unroll
        for (int t = 0; t < 8; ++t) {
            const int   iox = ioi[OIU[t]];
            const int   ioy = ioi[OIV[t]];
            const float fx  = frf[OIU[t]];
            const float fy  = frf[OIV[t]];
            const float w00 = (1.0f - fx) * (1.0f - fy);
            const float w01 = fx * (1.0f - fy);
            const float w10 = (1.0f - fx) * fy;
            const float w11 = fx * fy;
            const int base = (ly + HALO + ioy) * LSTR + (lx + HALO + iox);
            #pragma unroll
            for (int c = 0; c < Cd; ++c) {
                const float* p = &tile[c * PLANE + base];
                const float s = fmaf(p[0], w00,
                               fmaf(p[1], w01,
                               fmaf(p[LSTR], w10, p[LSTR + 1] * w11)));
                su[c] = fmaf(KUc[t], s, su[c]);
                sv[c] = fmaf(KVc[t], s, sv[c]);
            }
        }

        float o0 = 0.0f, o1 = 0.0f, o2 = 0.0f;
        #pragma unroll
        for (int c = 0; c < Cd; ++c) {
            const float a = su[c] * LUM[c];
            const float v = sv[c] * LUM[c];
            o0 = fmaf(a, a, o0);
            o1 = fmaf(v, v, o1);
            o2 = fmaf(a, v, o2);
        }

        const int gy = tileY + ly;
        const int gx = tileX + lx;
        const size_t plane = (size_t)Hd * Wd;
        const size_t base_o = (size_t)b * 3 * plane + (size_t)gy * Wd + gx;
        out[base_o]             = o0;
        out[base_o + plane]     = o1;
        out[base_o + 2 * plane] = o2;
    }
}

} // anonymous namespace

extern "C" void kernel_launch(void* const* d_in, const int* in_sizes, int n_in,
                              void* d_out, int out_size, void* d_ws, size_t ws_size,
                              hipStream_t stream)
{
    (void)in_sizes; (void)n_in; (void)out_size; (void)d_ws; (void)ws_size;
    const float* x     = (const float*)d_in[0];
    const float* sigma = (const float*)d_in[1];
    float* out = (float*)d_out;

    dim3 grid(Wd / TX, Hd / TY, Bd);   // 16 x 128 x 4
    dim3 block(TX, 4, 1);              // 256 threads = 8 wave32
    structure_tensor_kernel<<<grid, block, 0, stream>>>(x, sigma, out);
}